// GCNLayer_20547123544324
// MI455X (gfx1250) — compile-verified
//
#include <hip/hip_runtime.h>
#include <hip/hip_bf16.h>

typedef float v2f __attribute__((ext_vector_type(2)));
typedef float v8f __attribute__((ext_vector_type(8)));

#define D 64

// ---------------- degree count (int atomics, counter array is L2-resident) ----
__global__ __launch_bounds__(256) void gcn_degree(const int* __restrict__ dst,
                                                  int* __restrict__ cnt, int E) {
    int e = blockIdx.x * blockDim.x + threadIdx.x;
    if (e < E) atomicAdd(&cnt[dst[e]], 1);
}

// ---------------- dinv = rsqrt(deg + 1 self loop) ----------------------------
__global__ __launch_bounds__(256) void gcn_dinv(const int* __restrict__ cnt,
                                                float* __restrict__ dinv, int N) {
    int i = blockIdx.x * blockDim.x + threadIdx.x;
    if (i < N) dinv[i] = rsqrtf((float)(cnt[i] + 1));
}

// ---------------- xw = x @ W via V_WMMA_F32_16X16X4_F32 ----------------------
// One wave computes a 16x64 output tile: 4 accumulators, A loaded once per K-step.
__global__ __launch_bounds__(256) void gcn_gemm_wmma(const float* __restrict__ x,
                                                     const float* __restrict__ W,
                                                     float* __restrict__ xw, int N) {
    const int lane = threadIdx.x & 31;
    const int wave = blockIdx.x * (blockDim.x >> 5) + (threadIdx.x >> 5);
    const int row0 = wave * 16;
    if (row0 >= N) return;                 // whole-wave uniform exit (EXEC stays full)
    const int hi  = lane >> 4;             // 0: lanes 0-15, 1: lanes 16-31
    const int l15 = lane & 15;

    // clamp for robustness if N not a multiple of 16 (loads only; stores guarded)
    int arow = row0 + l15; if (arow >= N) arow = N - 1;

    v8f acc0 = {}, acc1 = {}, acc2 = {}, acc3 = {};

    for (int k0 = 0; k0 < D; k0 += 4) {
        // A 16x4: VGPR0 = K=k0+2*hi, VGPR1 = K=k0+2*hi+1  (per ISA layout table)
        const float* xp = x + (size_t)arow * D + k0 + hi * 2;
        v2f a; a.x = xp[0]; a.y = xp[1];

        // B 4x16 per 16-col group: VGPR0 row k0+2*hi, VGPR1 row k0+2*hi+1
        const float* wp0 = W + (size_t)(k0 + hi * 2) * D + l15;
        const float* wp1 = wp0 + D;
        v2f b0, b1, b2, b3;
        b0.x = wp0[0];  b0.y = wp1[0];
        b1.x = wp0[16]; b1.y = wp1[16];
        b2.x = wp0[32]; b2.y = wp1[32];
        b3.x = wp0[48]; b3.y = wp1[48];

        acc0 = __builtin_amdgcn_wmma_f32_16x16x4_f32(false, a, false, b0, (short)0, acc0, false, false);
        acc1 = __builtin_amdgcn_wmma_f32_16x16x4_f32(false, a, false, b1, (short)0, acc1, false, false);
        acc2 = __builtin_amdgcn_wmma_f32_16x16x4_f32(false, a, false, b2, (short)0, acc2, false, false);
        acc3 = __builtin_amdgcn_wmma_f32_16x16x4_f32(false, a, false, b3, (short)0, acc3, false, false);
    }

    // C/D layout: VGPR r -> row = row0 + r (+8 for upper lane half), col = l15
    #pragma unroll
    for (int r = 0; r < 8; ++r) {
        int row = row0 + r + hi * 8;
        if (row < N) {
            float* op = xw + (size_t)row * D + l15;
            op[0]  = acc0[r];
            op[16] = acc1[r];
            op[32] = acc2[r];
            op[48] = acc3[r];
        }
    }
}

// ---------------- out = bias + self-loop term --------------------------------
__global__ __launch_bounds__(256) void gcn_init_out(const float* __restrict__ xw,
                                                    const float* __restrict__ dinv,
                                                    const float* __restrict__ bias,
                                                    float* __restrict__ out, int N) {
    int i = blockIdx.x * blockDim.x + threadIdx.x;
    if (i >= N * D) return;
    int node = i >> 6, dch = i & (D - 1);
    float di = dinv[node];
    out[i] = bias[dch] + xw[i] * di * di;
}

// ---------------- edge scatter: 1 wave per edge, float2 per lane -------------
// Gather (256B coalesced, L2-resident) then 2x global_atomic_add_f32 per lane.
__global__ __launch_bounds__(256) void gcn_edge_scatter(const int* __restrict__ src,
                                                        const int* __restrict__ dst,
                                                        const float* __restrict__ xw,
                                                        const float* __restrict__ dinv,
                                                        float* __restrict__ out, int E) {
    int e = blockIdx.x * (blockDim.x >> 5) + (threadIdx.x >> 5);
    if (e >= E) return;
    int lane = threadIdx.x & 31;
    int s = src[e];
    int d = dst[e];
    float nrm = dinv[s] * dinv[d];
    float2 v = *((const float2*)(xw + (size_t)s * D) + lane);
    float* op = out + (size_t)d * D + lane * 2;
    unsafeAtomicAdd(op,     v.x * nrm);   // lowers to global_atomic_add_f32
    unsafeAtomicAdd(op + 1, v.y * nrm);
}

extern "C" void kernel_launch(void* const* d_in, const int* in_sizes, int n_in,
                              void* d_out, int out_size, void* d_ws, size_t ws_size,
                              hipStream_t stream) {
    const float* x  = (const float*)d_in[0];
    const int*   ei = (const int*)d_in[1];     // [2, E]: row0=src, row1=dst
    const float* W  = (const float*)d_in[2];
    const float* b  = (const float*)d_in[3];
    float* out = (float*)d_out;

    const int N = in_sizes[0] / D;
    const int E = in_sizes[1] / 2;
    const int* src = ei;
    const int* dst = ei + E;

    // workspace layout: [xw: N*64 f32][dinv: N f32][cnt: N i32]  (~26.4 MB)
    float* xw   = (float*)d_ws;
    float* dinv = xw + (size_t)N * D;
    int*   cnt  = (int*)(dinv + N);

    hipMemsetAsync(cnt, 0, (size_t)N * sizeof(int), stream);

    gcn_degree<<<(E + 255) / 256, 256, 0, stream>>>(dst, cnt, E);
    gcn_dinv<<<(N + 255) / 256, 256, 0, stream>>>(cnt, dinv, N);

    int rowTiles = (N + 15) / 16;                 // one wave per 16x64 tile
    gcn_gemm_wmma<<<(rowTiles + 7) / 8, 256, 0, stream>>>(x, W, xw, N);

    gcn_init_out<<<((size_t)N * D + 255) / 256, 256, 0, stream>>>(xw, dinv, b, out, N);

    gcn_edge_scatter<<<(E + 7) / 8, 256, 0, stream>>>(src, dst, xw, dinv, out, E);
}